// Model_72748156060317
// MI455X (gfx1250) — compile-verified
//
#include <hip/hip_runtime.h>
#include <math.h>

// Model constants (E and N taken from harness sizes at launch)
constexpr int kH    = 128;   // hidden
constexpr int kB    = 32;    // batch == wave32 lanes
constexpr int kNREL = 400;
constexpr int kRSZ  = 200;
constexpr float kTAU = 10.0f;

typedef __attribute__((ext_vector_type(2))) float v2f;
typedef __attribute__((ext_vector_type(8))) float v8f;

__device__ __forceinline__ float sigm(float x) { return 1.0f / (1.0f + expf(-x)); }

// ---------------------------------------------------------------------------
// Kernel 1: f32-WMMA LSTM (4 steps) + attention softmaxes + edge-weight
// softmaxes. One workgroup (8 waves). Wave w owns h-columns [16w,16w+16) for
// both 16-row batch tiles; for each it computes the i/f/g/o 16x16 tiles as
// WMMA fragments (K=128 -> 32 x V_WMMA_F32_16X16X4_F32 per operand matrix),
// applies the gate nonlinearities lane-locally, and keeps c in registers.
// ---------------------------------------------------------------------------
__global__ __launch_bounds__(256) void lstm_att_kernel(
    const int* __restrict__ input_r,
    const float* __restrict__ emb_W,
    const float* __restrict__ W_ih, const float* __restrict__ W_hh,
    const float* __restrict__ b_ih, const float* __restrict__ b_hh,
    const float* __restrict__ lin_W, const float* __restrict__ lin_b,
    float* __restrict__ att,    // [4][B][4]
    float* __restrict__ wsoft)  // [3][NREL][B]
{
    __shared__ float rnn[4][kB][kH];   // 64 KB LDS: LSTM hidden states
    const int tid  = threadIdx.x;
    const int lane = tid & 31;
    const int wid  = tid >> 5;     // 0..7
    const int hi   = lane >> 4;    // lane half (K-pair select for A/B frags)
    const int lo   = lane & 15;    // row (A) / col (B,C,D) within tile

    // c carried in C/D-fragment layout: elem r -> (m = 16*mt + r + 8*hi, col = 16*wid + lo)
    v8f cfrag[2];
    for (int mt = 0; mt < 2; ++mt)
        for (int r = 0; r < 8; ++r) cfrag[mt][r] = 0.0f;

    float bias[4];
    for (int g = 0; g < 4; ++g) {
        const int n = g * kH + wid * 16 + lo;   // gate column (per lane)
        bias[g] = b_ih[n] + b_hh[n];
    }

    for (int t = 0; t < 4; ++t) {
        for (int mt = 0; mt < 2; ++mt) {
            const int m = mt * 16 + lo;  // batch row this lane feeds into A
            const float* xrow = (t < 3) ? (emb_W + (size_t)input_r[m] * kH)
                                        : (emb_W + (size_t)kNREL * kH);
            v8f acc[4];
            for (int g = 0; g < 4; ++g)
                for (int r = 0; r < 8; ++r) acc[g][r] = 0.0f;

            // x_t @ W_ih^T
            for (int k0 = 0; k0 < kH; k0 += 4) {
                const int kk = k0 + 2 * hi;
                v2f a; a.x = xrow[kk]; a.y = xrow[kk + 1];
                for (int g = 0; g < 4; ++g) {
                    const int n = g * kH + wid * 16 + lo;
                    v2f bb; bb.x = W_ih[(size_t)n * kH + kk];
                    bb.y = W_ih[(size_t)n * kH + kk + 1];
                    acc[g] = __builtin_amdgcn_wmma_f32_16x16x4_f32(
                        false, a, false, bb, (short)0, acc[g], false, false);
                }
            }
            // h_{t-1} @ W_hh^T  (h_0 == 0 -> skip at t==0)
            if (t > 0) {
                for (int k0 = 0; k0 < kH; k0 += 4) {
                    const int kk = k0 + 2 * hi;
                    v2f a; a.x = rnn[t - 1][m][kk]; a.y = rnn[t - 1][m][kk + 1];
                    for (int g = 0; g < 4; ++g) {
                        const int n = g * kH + wid * 16 + lo;
                        v2f bb; bb.x = W_hh[(size_t)n * kH + kk];
                        bb.y = W_hh[(size_t)n * kH + kk + 1];
                        acc[g] = __builtin_amdgcn_wmma_f32_16x16x4_f32(
                            false, a, false, bb, (short)0, acc[g], false, false);
                    }
                }
            }
            // gate nonlinearities directly on fragments
            for (int r = 0; r < 8; ++r) {
                const float iv = sigm(acc[0][r] + bias[0]);
                const float fv = sigm(acc[1][r] + bias[1]);
                const float gv = tanhf(acc[2][r] + bias[2]);
                const float ov = sigm(acc[3][r] + bias[3]);
                const float cv = fv * cfrag[mt][r] + iv * gv;
                cfrag[mt][r] = cv;
                rnn[t][mt * 16 + r + 8 * hi][wid * 16 + lo] = ov * tanhf(cv);
            }
        }
        __syncthreads();
    }

    // Attention softmaxes: att[t][b][j] = softmax_j( rnn[t][b] . rnn[j][b] ), j<=t
    if (tid < 128) {
        const int b = tid >> 2, t = tid & 3;
        float ex[4]; float mx = -1e30f;
        for (int j = 0; j <= t; ++j) {
            float d = 0.0f;
            for (int k = 0; k < kH; ++k) d += rnn[t][b][k] * rnn[j][b][k];
            ex[j] = d; if (d > mx) mx = d;
        }
        float se = 0.0f;
        for (int j = 0; j <= t; ++j) { ex[j] = expf(ex[j] - mx); se += ex[j]; }
        const float inv = 1.0f / se;
        for (int j = 0; j <= t; ++j) att[t * (kB * 4) + b * 4 + j] = ex[j] * inv;
    }

    // Edge-weight softmax: wsoft[t][r][b] = softmax_r( TAU * (rnn[t][b].lin_W[r] + lin_b[r]) )
    // One wave per (t,b) group; lanes stripe r.
    for (int grp = wid; grp < 96; grp += 8) {
        const int t = grp >> 5, b = grp & 31;
        float v[13]; float mx = -1e30f;
        for (int q = 0; q < 13; ++q) {
            const int r = lane + q * 32;
            float val = -1e30f;
            if (r < kNREL) {
                float d = lin_b[r];
                for (int k = 0; k < kH; ++k) d += rnn[t][b][k] * lin_W[(size_t)r * kH + k];
                val = kTAU * d;
            }
            v[q] = val; if (val > mx) mx = val;
        }
        for (int off = 16; off > 0; off >>= 1) mx = fmaxf(mx, __shfl_xor(mx, off, 32));
        float se = 0.0f;
        for (int q = 0; q < 13; ++q) {
            const int r = lane + q * 32;
            if (r < kNREL) { v[q] = expf(v[q] - mx); se += v[q]; }
        }
        for (int off = 16; off > 0; off >>= 1) se += __shfl_xor(se, off, 32);
        const float inv = 1.0f / se;
        for (int q = 0; q < 13; ++q) {
            const int r = lane + q * 32;
            if (r < kNREL) wsoft[(size_t)t * kNREL * kB + (size_t)r * kB + b] = v[q] * inv;
        }
    }
}

// ---------------------------------------------------------------------------
// Dense tensors stored TRANSPOSED: X[e*32 + b]  (lane == batch)
// ---------------------------------------------------------------------------

// inp = att[t][b,1]*S1 + att[t][b,2]*S2 + onehot(input_x[b]) * att[t][b,0]
__global__ __launch_bounds__(256) void combine_kernel(
    const float* __restrict__ S1, const float* __restrict__ S2,
    const float* __restrict__ attT, const int* __restrict__ input_x,
    float* __restrict__ inp, int nS, int n)
{
    const int i = blockIdx.x * blockDim.x + threadIdx.x;
    if (i >= n) return;
    const int b = i & 31, e = i >> 5;
    float v = 0.0f;
    if (nS >= 1) v += attT[b * 4 + 1] * S1[i];
    if (nS >= 2) v += attT[b * 4 + 2] * S2[i];
    if (e == input_x[b]) v += attT[b * 4 + 0];
    inp[i] = v;
}

__global__ void zero_kernel(float* __restrict__ s, float* __restrict__ sums, int n)
{
    int i = blockIdx.x * blockDim.x + threadIdx.x;
    const int stride = gridDim.x * blockDim.x;
    for (; i < n; i += stride) s[i] = 0.0f;
    if (blockIdx.x == 0 && threadIdx.x < 32) sums[threadIdx.x] = 0.0f;
}

// One wave per edge; lane == batch. Edge data wave-uniform -> scalar loads;
// gathers/atomics hit 32 consecutive floats (single L2 line). All L2-resident.
__global__ __launch_bounds__(256) void propagate_kernel(
    const float* __restrict__ inp, const int* __restrict__ heads,
    const int* __restrict__ tails, const int* __restrict__ rels,
    const float* __restrict__ wT, float* __restrict__ s, int nE)
{
    const int e = blockIdx.x * 8 + (threadIdx.x >> 5);
    if (e >= nE) return;
    const int b = threadIdx.x & 31;
    const int hd = heads[e], tl = tails[e], rl = rels[e];
    const float w1 = wT[(size_t)rl * kB + b];
    const float w2 = wT[(size_t)(rl + kRSZ) * kB + b];
    const float xh = inp[(size_t)hd * kB + b];
    const float xt = inp[(size_t)tl * kB + b];
    unsafeAtomicAdd(&s[(size_t)tl * kB + b], xh * w1);   // s_h scatter to tails
    unsafeAtomicAdd(&s[(size_t)hd * kB + b], xt * w2);   // s_t scatter to heads
}

__global__ __launch_bounds__(256) void colsum_kernel(
    const float* __restrict__ s, float* __restrict__ sums, int E_)
{
    __shared__ float red[256];
    const int b = threadIdx.x & 31;
    const int w = threadIdx.x >> 5;
    float acc = 0.0f;
    for (int e = blockIdx.x * 8 + w; e < E_; e += gridDim.x * 8)
        acc += s[(size_t)e * kB + b];
    red[threadIdx.x] = acc;
    __syncthreads();
    if (threadIdx.x < 32) {
        float t = 0.0f;
        for (int q = 0; q < 8; ++q) t += red[q * 32 + threadIdx.x];
        unsafeAtomicAdd(&sums[threadIdx.x], t);
    }
}

__global__ __launch_bounds__(256) void scale_kernel(
    float* __restrict__ s, const float* __restrict__ sums, int n)
{
    const int i = blockIdx.x * blockDim.x + threadIdx.x;
    if (i >= n) return;
    const int b = i & 31;
    s[i] *= 1.0f / fmaxf(sums[b], 1e-7f);
}

// t==T: output = att-combination only, written in (B,E) order.
__global__ __launch_bounds__(256) void final_kernel(
    const float* __restrict__ S1, const float* __restrict__ S2,
    const float* __restrict__ S3, const float* __restrict__ attT,
    const int* __restrict__ input_x, float* __restrict__ out, int E_, int n)
{
    const int i = blockIdx.x * blockDim.x + threadIdx.x;
    if (i >= n) return;
    const int b = i / E_, e = i - b * E_;
    const size_t j = (size_t)e * kB + b;
    float v = attT[b * 4 + 1] * S1[j] + attT[b * 4 + 2] * S2[j] + attT[b * 4 + 3] * S3[j];
    if (e == input_x[b]) v += attT[b * 4 + 0];
    out[i] = v;
}

// ---------------------------------------------------------------------------
extern "C" void kernel_launch(void* const* d_in, const int* in_sizes, int n_in,
                              void* d_out, int out_size, void* d_ws, size_t ws_size,
                              hipStream_t stream)
{
    const int*   input_x = (const int*)d_in[0];
    const int*   input_r = (const int*)d_in[1];
    const int*   heads   = (const int*)d_in[2];
    const int*   tails   = (const int*)d_in[3];
    const int*   rels    = (const int*)d_in[4];
    const float* emb_W   = (const float*)d_in[5];
    const float* W_ih    = (const float*)d_in[6];
    const float* W_hh    = (const float*)d_in[7];
    const float* b_ih    = (const float*)d_in[8];
    const float* b_hh    = (const float*)d_in[9];
    const float* lin_W   = (const float*)d_in[10];
    const float* lin_b   = (const float*)d_in[11];
    float* out = (float*)d_out;

    const int nE = in_sizes[2];          // 1,000,000 edges
    const int E_ = out_size / kB;        // 40,000 entities
    const size_t EB = (size_t)E_ * kB;

    // workspace layout (floats): S1,S2,S3 | inp | att | wsoft | sums  (~21 MB)
    float* ws    = (float*)d_ws;
    float* S[3]  = { ws, ws + EB, ws + 2 * EB };
    float* inp   = ws + 3 * EB;
    float* att   = ws + 4 * EB;                 // 4*32*4
    float* wsoft = att + 4 * kB * 4;            // 3*400*32
    float* sums  = wsoft + 3 * kNREL * kB;      // 32

    lstm_att_kernel<<<1, 256, 0, stream>>>(input_r, emb_W, W_ih, W_hh,
                                           b_ih, b_hh, lin_W, lin_b, att, wsoft);

    const int n = (int)EB;
    const int blocksEB = (n + 255) / 256;
    for (int t = 0; t < 3; ++t) {
        combine_kernel<<<blocksEB, 256, 0, stream>>>(S[0], S[1], att + t * kB * 4,
                                                     input_x, inp, t, n);
        zero_kernel<<<blocksEB, 256, 0, stream>>>(S[t], sums, n);
        propagate_kernel<<<(nE + 7) / 8, 256, 0, stream>>>(
            inp, heads, tails, rels, wsoft + (size_t)t * kNREL * kB, S[t], nE);
        colsum_kernel<<<1024, 256, 0, stream>>>(S[t], sums, E_);
        scale_kernel<<<blocksEB, 256, 0, stream>>>(S[t], sums, n);
    }
    final_kernel<<<(out_size + 255) / 256, 256, 0, stream>>>(
        S[0], S[1], S[2], att + 3 * kB * 4, input_x, out, E_, out_size);
}